// CTC_56264071577713
// MI455X (gfx1250) — compile-verified
//
#include <hip/hip_runtime.h>
#include <hip/hip_bf16.h>
#include <math.h>

// Problem constants (from reference setup_inputs)
#define BB   16
#define TT   800
#define DD   512
#define VV   4000
#define LMAX 100
#define SS   201      // 2*LMAX + 1 extended states
#define SC   112      // padded column count for emit buffer (1 blank + 100 labels + 11 pad)
#define LDA  516      // LDS row stride: 516 mod 64 = 4 -> 16 distinct banks, 16B-aligned rows
#define NEGF (-1e30f)

typedef float v2f __attribute__((ext_vector_type(2)));
typedef float v8f __attribute__((ext_vector_type(8)));

__device__ __forceinline__ float logaddexp_f(float a, float b) {
    float m  = fmaxf(a, b);
    float mn = fminf(a, b);
    return m + log1pf(__expf(mn - m));
}

// Merge per-lane streaming logsumexp states (m,s) across the 16 lanes of a
// lane-half via butterfly (xor 1,2,4,8 never crosses the half on wave32).
__device__ __forceinline__ void lse_merge16(float& m, float& s) {
#pragma unroll
    for (int d = 1; d < 16; d <<= 1) {
        float om = __shfl_xor(m, d, 32);
        float os = __shfl_xor(s, d, 32);
        float nm = fmaxf(m, om);
        s = s * __expf(m - nm) + os * __expf(om - nm);
        m = nm;
    }
}

// Stage a 16x512 f32 tile of hs into LDS (row stride LDA) using the CDNA5
// async global->LDS DMA path. 256 threads x 8 x B128 = 32KB. Caller barriers.
__device__ __forceinline__ void stage_A_async(const float* __restrict__ hs, int row0,
                                              float* sA, int tid) {
#pragma unroll
    for (int j = 0; j < 8; ++j) {
        const int c  = tid + 256 * j;      // 0..2047 float4 chunks
        const int r  = c >> 7;             // /128 -> row 0..15
        const int kq = c & 127;            // float4 index within row
        uint32_t loff = (uint32_t)(uintptr_t)(&sA[r * LDA + kq * 4]);
        const float* gp = hs + (size_t)(row0 + r) * DD + kq * 4;
        asm volatile("global_load_async_to_lds_b128 %0, %1, off"
                     :: "v"(loff), "v"(gp) : "memory");
    }
    asm volatile("s_wait_asynccnt 0x0" ::: "memory");
}

// ---------------------------------------------------------------------------
// Kernel 1: Z[r] = logsumexp_v( hs[r,:] @ W[:,v] + bias[v] )   r = b*T + t
// 800 blocks x 256 threads (8 waves). Each block: 16-row tile, waves split V.
// ---------------------------------------------------------------------------
__global__ __launch_bounds__(256)
void ctc_logz_kernel(const float* __restrict__ hs, const float* __restrict__ W,
                     const float* __restrict__ bias, float* __restrict__ Z) {
    __shared__ float sA[16 * LDA];
    __shared__ float redM[8 * 16];
    __shared__ float redS[8 * 16];

    const int tid  = threadIdx.x;
    const int wave = tid >> 5;
    const int lane = tid & 31;
    const int half = lane >> 4;
    const int l16  = lane & 15;
    const int row0 = blockIdx.x * 16;

    stage_A_async(hs, row0, sA, tid);
    __syncthreads();

    // Per-lane streaming logsumexp state per C-row (acc[i] -> row i + 8*half).
    // Each lane folds only its own columns; merged across lanes once at end.
    float rm[8], rs[8];
#pragma unroll
    for (int i = 0; i < 8; ++i) { rm[i] = NEGF; rs[i] = 0.f; }

    const float* aBase = &sA[l16 * LDA + 2 * half];

    for (int nt = wave; nt < VV / 16; nt += 8) {
        const int n = nt * 16 + l16;
        v8f acc = {0.f, 0.f, 0.f, 0.f, 0.f, 0.f, 0.f, 0.f};
        const float* wp = W + (size_t)(2 * half) * VV + n;
#pragma unroll 4
        for (int k0 = 0; k0 < DD; k0 += 4) {
            v2f a;  a.x  = aBase[k0];  a.y  = aBase[k0 + 1];
            v2f bm; bm.x = wp[0];      bm.y = wp[VV];
            wp += 4 * VV;
            acc = __builtin_amdgcn_wmma_f32_16x16x4_f32(
                false, a, false, bm, (short)0, acc, false, false);
        }
        const float bv = bias[n];
#pragma unroll
        for (int i = 0; i < 8; ++i) {
            float x  = acc[i] + bv;
            float nm = fmaxf(rm[i], x);
            rs[i] = rs[i] * __expf(rm[i] - nm) + __expf(x - nm);
            rm[i] = nm;
        }
    }

    // One butterfly merge per row at wave end (no bpermute in the hot loop).
#pragma unroll
    for (int i = 0; i < 8; ++i) {
        lse_merge16(rm[i], rs[i]);
        if (l16 == 0) {
            redM[wave * 16 + half * 8 + i] = rm[i];
            redS[wave * 16 + half * 8 + i] = rs[i];
        }
    }
    __syncthreads();

    if (tid < 16) {
        float m = NEGF, s = 0.f;
        for (int w = 0; w < 8; ++w) {
            float pm = redM[w * 16 + tid];
            float ps = redS[w * 16 + tid];
            float nm = fmaxf(m, pm);
            s = s * __expf(m - nm) + ps * __expf(pm - nm);
            m = nm;
        }
        Z[row0 + tid] = m + __logf(s);
    }
}

// ---------------------------------------------------------------------------
// Kernel 2: emit[r][c] = hs[r,:] @ W[:, col_b[c]] + bias[col] - Z[r]
// col_b = {0(blank), ys_pad[b,0..99], 0-pad...} ; 800 blocks of 16 rows
// (each block stays inside one batch because T=800 is a multiple of 16).
// ---------------------------------------------------------------------------
__global__ __launch_bounds__(256)
void ctc_emit_kernel(const float* __restrict__ hs, const float* __restrict__ W,
                     const float* __restrict__ bias, const int* __restrict__ ys_pad,
                     const float* __restrict__ Z, float* __restrict__ emit) {
    __shared__ float sA[16 * LDA];
    __shared__ int   sCol[SC];

    const int tid  = threadIdx.x;
    const int wave = tid >> 5;
    const int lane = tid & 31;
    const int half = lane >> 4;
    const int l16  = lane & 15;
    const int row0 = blockIdx.x * 16;
    const int b    = row0 / TT;

    stage_A_async(hs, row0, sA, tid);
    if (tid < SC) {
        int c = 0;
        if (tid >= 1 && tid <= LMAX) c = ys_pad[b * LMAX + (tid - 1)];
        sCol[tid] = c;
    }
    __syncthreads();

    if (wave < 7) {   // 7 waves x 16 = 112 columns
        const int ci = wave * 16 + l16;
        const int n  = sCol[ci];
        const float* aBase = &sA[l16 * LDA + 2 * half];
        v8f acc = {0.f, 0.f, 0.f, 0.f, 0.f, 0.f, 0.f, 0.f};
        const float* wp = W + (size_t)(2 * half) * VV + n;
#pragma unroll 4
        for (int k0 = 0; k0 < DD; k0 += 4) {
            v2f a;  a.x  = aBase[k0];  a.y  = aBase[k0 + 1];
            v2f bm; bm.x = wp[0];      bm.y = wp[VV];
            wp += 4 * VV;
            acc = __builtin_amdgcn_wmma_f32_16x16x4_f32(
                false, a, false, bm, (short)0, acc, false, false);
        }
        const float bv = bias[n];
#pragma unroll
        for (int i = 0; i < 8; ++i) {
            const int r = row0 + half * 8 + i;
            emit[(size_t)r * SC + ci] = acc[i] + bv - Z[r];
        }
    }
}

// ---------------------------------------------------------------------------
// Kernel 3: CTC alpha recursion. One block per batch, thread s = state.
// ---------------------------------------------------------------------------
__global__ __launch_bounds__(256)
void ctc_alpha_kernel(const float* __restrict__ emit, const int* __restrict__ hlens,
                      const int* __restrict__ ys_pad, const int* __restrict__ ys_lens,
                      float* __restrict__ lossPartial) {
    __shared__ float al[SS];

    const int b = blockIdx.x;
    const int s = threadIdx.x;
    const bool valid = (s < SS);
    const int hlen = hlens[b];
    const float* em = emit + (size_t)b * TT * SC;

    // Per-state constants
    int  eidx = 0;
    bool skip = false;
    if (valid && (s & 1)) {
        const int li = (s - 1) >> 1;
        eidx = 1 + li;
        if (s >= 2) {
            const int lab  = ys_pad[b * LMAX + li];
            const int lab2 = (li >= 1) ? ys_pad[b * LMAX + li - 1] : 0;
            skip = (lab != 0) && (lab != lab2);
        }
    }

    if (valid) al[s] = (s < 2) ? em[s] : NEGF;   // t=0 init: eidx==s for s<2
    __syncthreads();

    for (int t = 1; t < hlen; ++t) {
        float a  = valid ? al[s] : NEGF;
        float a1 = (valid && s >= 1) ? al[s - 1] : NEGF;
        float a2 = (valid && skip)   ? al[s - 2] : NEGF;
        __syncthreads();
        if (valid) {
            float l = logaddexp_f(logaddexp_f(a, a1), a2);
            al[s] = l + em[(size_t)t * SC + eidx];
        }
        __syncthreads();
    }

    if (s == 0) {
        const int idx = 2 * ys_lens[b];
        float loss = -logaddexp_f(al[idx], al[idx - 1]);
        bool ok = (loss == loss) && (loss < 1e29f) && (loss > -3.0e38f);
        lossPartial[b] = ok ? loss : 0.f;
    }
}

// ---------------------------------------------------------------------------
// Kernel 4: final reduction  out = sum(loss_b) / B
// ---------------------------------------------------------------------------
__global__ void ctc_sum_kernel(const float* __restrict__ lossPartial, float* __restrict__ out) {
    if (threadIdx.x == 0 && blockIdx.x == 0) {
        float s = 0.f;
        for (int i = 0; i < BB; ++i) s += lossPartial[i];
        out[0] = s / (float)BB;
    }
}

// ---------------------------------------------------------------------------
extern "C" void kernel_launch(void* const* d_in, const int* in_sizes, int n_in,
                              void* d_out, int out_size, void* d_ws, size_t ws_size,
                              hipStream_t stream) {
    const float* hs      = (const float*)d_in[0];   // [B,T,D]
    const float* W       = (const float*)d_in[1];   // [D,V]
    const float* bias    = (const float*)d_in[2];   // [V]
    const int*   hlens   = (const int*)d_in[3];     // [B]
    const int*   ys_pad  = (const int*)d_in[4];     // [B,Lmax]
    const int*   ys_lens = (const int*)d_in[5];     // [B]
    float* out = (float*)d_out;

    char* ws = (char*)d_ws;
    float* Z           = (float*)(ws);                                  // 12800 floats
    float* emit        = (float*)(ws + 51200);                          // 12800*112 floats
    float* lossPartial = (float*)(ws + 51200 + (size_t)BB * TT * SC * 4);

    const int nRowTiles = (BB * TT) / 16;   // 800

    ctc_logz_kernel <<<nRowTiles, 256, 0, stream>>>(hs, W, bias, Z);
    ctc_emit_kernel <<<nRowTiles, 256, 0, stream>>>(hs, W, bias, ys_pad, Z, emit);
    ctc_alpha_kernel<<<BB,        256, 0, stream>>>(emit, hlens, ys_pad, ys_lens, lossPartial);
    ctc_sum_kernel  <<<1,          32, 0, stream>>>(lossPartial, out);
}